// FCAdaINLayer_68719477052
// MI455X (gfx1250) — compile-verified
//
#include <hip/hip_runtime.h>
#include <hip/hip_bf16.h>

typedef __attribute__((ext_vector_type(2))) float v2f;
typedef __attribute__((ext_vector_type(8))) float v8f;

#define IN_C   64
#define OUT_C  64
#define FEAT_C 128

// ---------------------------------------------------------------------------
// Pass 1/3: h = x @ W_fc + b_fc via V_WMMA_F32_16X16X4_F32.
// Each wave owns a 16-row tile: A = x[16 x 64], B = W_fc[64 x 64] (in regs),
// C/D = 4 tiles of 16x16 f32 accumulators.
// MODE 0: atomically accumulate per-(group,channel) sum(h), sum(h^2), count.
// MODE 1: out = relu(h * scale[g,c] + bias[g,c]).
// ---------------------------------------------------------------------------
template <int MODE>
__global__ __launch_bounds__(256) void fcadain_gemm_pass(
    const float* __restrict__ x,      // [N, 64]
    const float* __restrict__ Wfc,    // [64, 64]
    const float* __restrict__ bfc,    // [64]
    const int*   __restrict__ seg,    // [N]
    float* __restrict__ sum_h,        // [G, 64]   (MODE 1: scale)
    float* __restrict__ sum_h2,       // [G, 64]   (MODE 1: bias)
    float* __restrict__ cnt,          // [G]
    float* __restrict__ out,          // [N, 64]
    int N, int nTiles)
{
    const int lane = threadIdx.x & 31;
    const int wave = threadIdx.x >> 5;
    const int m    = lane & 15;   // row-in-tile (A) / col-in-tile (B,C,D)
    const int hi   = lane >> 4;   // half-wave select

    // --- Preload all B fragments of W_fc into registers (64 x v2f = 128 VGPR).
    // ISA 32-bit 4x16 B layout (mirror of A 16x4): VGPR0 holds K=k0(+2*hi),
    // VGPR1 holds K+1, lanes = N.
    v2f bf[64];
#pragma unroll
    for (int j = 0; j < 4; ++j) {
#pragma unroll
        for (int k = 0; k < 16; ++k) {
            const int kk  = 4 * k + 2 * hi;
            const int col = j * 16 + m;
            v2f b;
            b.x = Wfc[kk * OUT_C + col];
            b.y = Wfc[(kk + 1) * OUT_C + col];
            bf[j * 16 + k] = b;
        }
    }
    float bv[4];
#pragma unroll
    for (int j = 0; j < 4; ++j) bv[j] = bfc[j * 16 + m];

    for (int tile = blockIdx.x * 8 + wave; tile < nTiles; tile += gridDim.x * 8) {
        const int rowBase = tile * 16;

        // A fragment source row for this lane (clamped for ragged tail).
        int arow = rowBase + m;
        if (arow >= N) arow = N - 1;
        const float* xr = x + (size_t)arow * IN_C;

        v8f acc[4];
#pragma unroll
        for (int j = 0; j < 4; ++j) acc[j] = (v8f){0.f,0.f,0.f,0.f,0.f,0.f,0.f,0.f};

        // 16 K-steps of 4; A fragment reused across the 4 N-tiles.
#pragma unroll
        for (int k = 0; k < 16; ++k) {
            const v2f a = *(const v2f*)(xr + 4 * k + 2 * hi);
#pragma unroll
            for (int j = 0; j < 4; ++j) {
                acc[j] = __builtin_amdgcn_wmma_f32_16x16x4_f32(
                    /*neg_a=*/false, a, /*neg_b=*/false, bf[j * 16 + k],
                    /*c_mod=*/(short)0, acc[j],
                    /*reuse_a=*/false, /*reuse_b=*/false);
            }
        }

        // Group id of row (rowBase + m), one per lane 0..15.
        int sidx = rowBase + m;
        if (sidx >= N) sidx = N - 1;
        const int sg = seg[sidx];

        // Epilogue. C/D layout: VGPR r -> row (r + 8*hi), lanes -> col.
#pragma unroll
        for (int j = 0; j < 4; ++j) {
            const int col = j * 16 + m;
#pragma unroll
            for (int r = 0; r < 8; ++r) {
                const float v   = acc[j][r] + bv[j];
                const int   glo = __builtin_amdgcn_readlane(sg, r);
                const int   ghi = __builtin_amdgcn_readlane(sg, r + 8);
                const int   g   = hi ? ghi : glo;
                const int   row = rowBase + r + 8 * hi;
                if (row < N) {
                    if (MODE == 0) {
                        atomicAdd(&sum_h [g * OUT_C + col], v);
                        atomicAdd(&sum_h2[g * OUT_C + col], v * v);
                    } else {
                        const float s = sum_h [g * OUT_C + col];  // scale
                        const float t = sum_h2[g * OUT_C + col];  // bias
                        float o = fmaf(v, s, t);
                        o = o > 0.f ? o : 0.f;
                        out[(size_t)row * OUT_C + col] = o;
                    }
                }
            }
        }
        if (MODE == 0) {
            if (lane < 16 && (rowBase + m) < N)
                atomicAdd(&cnt[sg], 1.0f);
        }
    }
}

// ---------------------------------------------------------------------------
// Pass 2: per (g,c) compute mu/sig (tiny GEMM) and fold stats into
// scale/bias, overwriting sum_h/sum_h2 in place.
// ---------------------------------------------------------------------------
__global__ __launch_bounds__(256) void fcadain_params(
    const float* __restrict__ feat,   // [G, 128]
    const float* __restrict__ Wmu,    // [128, 64]
    const float* __restrict__ bmu,    // [64]
    const float* __restrict__ Wsig,   // [128, 64]
    const float* __restrict__ bsig,   // [64]
    float* __restrict__ sum_h,        // in: sum(h)   out: scale
    float* __restrict__ sum_h2,       // in: sum(h^2) out: bias
    const float* __restrict__ cnt,    // [G]
    int G)
{
    const int t = blockIdx.x * 256 + threadIdx.x;
    if (t >= G * OUT_C) return;
    const int g = t >> 6;
    const int c = t & 63;

    const float* f = feat + (size_t)g * FEAT_C;
    float mu = bmu[c];
    float sg = bsig[c];
#pragma unroll 8
    for (int k = 0; k < FEAT_C; ++k) {
        const float fv = f[k];
        mu = fmaf(fv, Wmu [k * OUT_C + c], mu);
        sg = fmaf(fv, Wsig[k * OUT_C + c], sg);
    }

    float cn = cnt[g];
    cn = cn > 1.f ? cn : 1.f;
    const float mean = sum_h[t] / cn;
    float var = sum_h2[t] / cn - mean * mean;
    var = var > 0.f ? var : 0.f;
    const float inv   = rsqrtf(var + 1e-14f);
    const float scale = sg * inv;
    sum_h [t] = scale;
    sum_h2[t] = mu - mean * scale;
}

// ---------------------------------------------------------------------------
extern "C" void kernel_launch(void* const* d_in, const int* in_sizes, int n_in,
                              void* d_out, int out_size, void* d_ws, size_t ws_size,
                              hipStream_t stream) {
    const float* x    = (const float*)d_in[0];
    const float* feat = (const float*)d_in[1];
    const int*   seg  = (const int*)  d_in[2];
    // d_in[3] = num_groups scalar (value known via in_sizes)
    const float* Wfc  = (const float*)d_in[4];
    const float* bfc  = (const float*)d_in[5];
    const float* Wmu  = (const float*)d_in[6];
    const float* bmu  = (const float*)d_in[7];
    const float* Wsig = (const float*)d_in[8];
    const float* bsig = (const float*)d_in[9];
    float* out = (float*)d_out;

    const int N = in_sizes[2];            // latent_point_batch length
    const int G = in_sizes[1] / FEAT_C;   // origin_feat rows

    float* ws     = (float*)d_ws;
    float* sum_h  = ws;                       // G*64
    float* sum_h2 = ws + (size_t)G * OUT_C;   // G*64
    float* cnt    = ws + (size_t)2 * G * OUT_C; // G

    const size_t zbytes = ((size_t)2 * G * OUT_C + G) * sizeof(float);
    hipMemsetAsync(d_ws, 0, zbytes, stream);

    const int nTiles = (N + 15) / 16;
    int blocks = (nTiles + 7) / 8;
    if (blocks > 4096) blocks = 4096;

    fcadain_gemm_pass<0><<<blocks, 256, 0, stream>>>(
        x, Wfc, bfc, seg, sum_h, sum_h2, cnt, out, N, nTiles);

    fcadain_params<<<(G * OUT_C + 255) / 256, 256, 0, stream>>>(
        feat, Wmu, bmu, Wsig, bsig, sum_h, sum_h2, cnt, G);

    fcadain_gemm_pass<1><<<blocks, 256, 0, stream>>>(
        x, Wfc, bfc, seg, sum_h, sum_h2, cnt, out, N, nTiles);
}